// GraphSAGE_73529840107534
// MI455X (gfx1250) — compile-verified
//
#include <hip/hip_runtime.h>

typedef __attribute__((ext_vector_type(2))) float v2f;
typedef __attribute__((ext_vector_type(8))) float v8f;

#define FEAT 128
#define LN_EPS 1e-5f
#define SCAN_T 1024

// ---------------- utility kernels ----------------

__global__ void zero_i32(int* __restrict__ p, long n) {
    long i = (long)blockIdx.x * blockDim.x + threadIdx.x;
    long stride = (long)gridDim.x * blockDim.x;
    for (; i < n; i += stride) p[i] = 0;
}

__global__ void degree_count(const long long* __restrict__ ei, int* __restrict__ deg, int E) {
    int e = blockIdx.x * blockDim.x + threadIdx.x;
    if (e < E) atomicAdd(&deg[(int)ei[(long)E + e]], 1);
}

__global__ void inv_count(const int* __restrict__ deg, float* __restrict__ inv, int n) {
    int i = blockIdx.x * blockDim.x + threadIdx.x;
    if (i < n) {
        int d = deg[i];
        inv[i] = 1.0f / (float)(d > 1 ? d : 1);
    }
}

// Single-block exclusive scan of deg[0..N) -> offs[0..N], offs[N] = E.
// Each thread serially sums a contiguous chunk, LDS Hillis-Steele across
// the 1024 partials, then each thread emits its chunk's exclusive scan.
__global__ void __launch_bounds__(SCAN_T) exscan_deg(const int* __restrict__ deg,
                                                     int* __restrict__ offs, int N) {
    __shared__ int part[SCAN_T];
    const int tid = threadIdx.x;
    const int chunk = (N + SCAN_T - 1) / SCAN_T;
    const int b = tid * chunk;
    const int e = (b + chunk < N) ? (b + chunk) : N;
    int s = 0;
    for (int i = b; i < e; ++i) s += deg[i];
    part[tid] = s;
    __syncthreads();
    for (int off = 1; off < SCAN_T; off <<= 1) {
        int v = (tid >= off) ? part[tid - off] : 0;
        __syncthreads();
        part[tid] += v;
        __syncthreads();
    }
    int base = (tid == 0) ? 0 : part[tid - 1];
    for (int i = b; i < e; ++i) { offs[i] = base; base += deg[i]; }
    if (tid == SCAN_T - 1) offs[N] = base;   // == E
}

// Fill CSR buckets (order nondeterministic; canonicalized by sort_buckets).
__global__ void bucket_fill(const long long* __restrict__ ei, const int* __restrict__ offs,
                            int* __restrict__ cursor, int* __restrict__ srcs, int E) {
    int e = blockIdx.x * blockDim.x + threadIdx.x;
    if (e < E) {
        int d = (int)ei[(long)E + e];
        int pos = offs[d] + atomicAdd(&cursor[d], 1);
        srcs[pos] = (int)ei[e];
    }
}

// Canonicalize each bucket (ascending src). Ties are identical rows, so the
// later fp sum order becomes fully deterministic. One thread per node,
// insertion sort over avg-degree-16 buckets (one-time cost).
__global__ void sort_buckets(const int* __restrict__ offs, int* __restrict__ srcs, int N) {
    int n = blockIdx.x * blockDim.x + threadIdx.x;
    if (n >= N) return;
    int b = offs[n], e = offs[n + 1];
    for (int i = b + 1; i < e; ++i) {
        int v = srcs[i];
        int j = i - 1;
        while (j >= b && srcs[j] > v) { srcs[j + 1] = srcs[j]; --j; }
        srcs[j + 1] = v;
    }
}

// Mean-neighbor aggregation from CSR: one wave per destination node.
// Pure gather + streaming store; no atomics. Prefetch 4 edges ahead
// (global_prefetch_b8) to hide the random-row gather latency.
__global__ void csr_aggregate(const int* __restrict__ offs, const int* __restrict__ srcs,
                              const float* __restrict__ h, const float* __restrict__ invc,
                              float* __restrict__ agg, int N) {
    int node = blockIdx.x * (blockDim.x >> 5) + (threadIdx.x >> 5);
    int lane = threadIdx.x & 31;
    if (node >= N) return;
    const int beg = offs[node], end = offs[node + 1];
    float4 s = {0.f, 0.f, 0.f, 0.f};
    for (int i = beg; i < end; ++i) {
        if (i + 4 < end) {
            long ps = (long)srcs[i + 4];
            __builtin_prefetch(h + ps * FEAT + lane * 4, 0, 0);
        }
        long src = (long)srcs[i];
        const float4 v = *(const float4*)(h + src * FEAT + lane * 4);
        s.x += v.x; s.y += v.y; s.z += v.z; s.w += v.w;
    }
    const float inv = invc[node];
    s.x *= inv; s.y *= inv; s.z *= inv; s.w *= inv;
    *(float4*)(agg + (size_t)node * FEAT + lane * 4) = s;
}

// ---------------- fused dual-GEMM + bias + LayerNorm + ReLU ----------------
// Each wave: 16 nodes x 128 output features.
//   acc[t] (t=0..7) is a 16x16 fp32 WMMA C/D tile covering features [t*16, t*16+16).
//   K=128 consumed in 32 steps of V_WMMA_F32_16X16X4_F32 (exact fp32 math),
//   accumulating both lin_l(agg*inv_cnt) and lin_r(h) into the same tile.
__global__ void __launch_bounds__(256) gemm_ln_relu(
    const float* __restrict__ agg, const float* __restrict__ hin,
    const float* __restrict__ invc,
    const float* __restrict__ Wl, const float* __restrict__ bl,
    const float* __restrict__ Wr,
    const float* __restrict__ gamma, const float* __restrict__ beta,
    float* __restrict__ hout, int N)
{
    const int lane = threadIdx.x & 31;
    const int wave = threadIdx.x >> 5;
    const int m0 = (blockIdx.x * (blockDim.x >> 5) + wave) * 16;
    if (m0 >= N) return;   // wave-uniform: EXEC stays all-1s for WMMA

    const int l15 = lane & 15;
    const int hi  = lane >> 4;            // which half of the wave
    const int arow = m0 + l15;            // node supplying this lane's A fragment

    const float* ap = agg + (size_t)arow * FEAT + 2 * hi;
    const float* hp = hin + (size_t)arow * FEAT + 2 * hi;

    v8f acc[8];
#pragma unroll
    for (int t = 0; t < 8; ++t) acc[t] = (v8f){0.f,0.f,0.f,0.f,0.f,0.f,0.f,0.f};

#pragma unroll 2
    for (int k0 = 0; k0 < FEAT; k0 += 4) {
        // A fragments: A[l15, 2*hi .. 2*hi+1] for this K-slab
        v2f aA = *(const v2f*)(ap + k0);   // agg already carries inv_cnt scaling
        v2f aH = *(const v2f*)(hp + k0);
#pragma unroll
        for (int t = 0; t < 8; ++t) {
            const int oc = t * 16 + l15;  // output feature column = B's N index
            // B[k,n] = W[n,k]: lane n loads rows k0+2*hi, k0+2*hi+1 of column oc
            v2f bL = *(const v2f*)(Wl + (size_t)oc * FEAT + k0 + 2 * hi);
            v2f bR = *(const v2f*)(Wr + (size_t)oc * FEAT + k0 + 2 * hi);
            acc[t] = __builtin_amdgcn_wmma_f32_16x16x4_f32(
                false, aA, false, bL, (short)0, acc[t], false, false);
            acc[t] = __builtin_amdgcn_wmma_f32_16x16x4_f32(
                false, aH, false, bR, (short)0, acc[t], false, false);
        }
    }

    // bias (bl broadcast along nodes)
#pragma unroll
    for (int t = 0; t < 8; ++t) {
        const float bv = bl[t * 16 + l15];
#pragma unroll
        for (int r = 0; r < 8; ++r) acc[t][r] += bv;
    }

    // LayerNorm: node (m0 + r + 8*hi)'s 128 features live in 8 tile registers
    // across the 16 lanes of this half-wave -> per-lane sum over t, then
    // shfl_xor butterfly within width 16.
    float mean[8], rstd[8];
#pragma unroll
    for (int r = 0; r < 8; ++r) {
        float s = 0.f;
#pragma unroll
        for (int t = 0; t < 8; ++t) s += acc[t][r];
        s += __shfl_xor(s, 1, 16);
        s += __shfl_xor(s, 2, 16);
        s += __shfl_xor(s, 4, 16);
        s += __shfl_xor(s, 8, 16);
        mean[r] = s * (1.0f / FEAT);
    }
#pragma unroll
    for (int r = 0; r < 8; ++r) {
        float s = 0.f;
#pragma unroll
        for (int t = 0; t < 8; ++t) {
            float d = acc[t][r] - mean[r];
            s += d * d;
        }
        s += __shfl_xor(s, 1, 16);
        s += __shfl_xor(s, 2, 16);
        s += __shfl_xor(s, 4, 16);
        s += __shfl_xor(s, 8, 16);
        rstd[r] = rsqrtf(s * (1.0f / FEAT) + LN_EPS);
    }

    // scale/shift + ReLU + store
#pragma unroll
    for (int t = 0; t < 8; ++t) {
        const int feat = t * 16 + l15;
        const float g  = gamma[feat];
        const float be = beta[feat];
#pragma unroll
        for (int r = 0; r < 8; ++r) {
            const int node = m0 + r + 8 * hi;
            float y = (acc[t][r] - mean[r]) * rstd[r] * g + be;
            y = y > 0.f ? y : 0.f;
            hout[(size_t)node * FEAT + feat] = y;
        }
    }
}

// ---------------- host-side launch ----------------

extern "C" void kernel_launch(void* const* d_in, const int* in_sizes, int n_in,
                              void* d_out, int out_size, void* d_ws, size_t ws_size,
                              hipStream_t stream) {
    const float*     x     = (const float*)d_in[0];
    const long long* ei    = (const long long*)d_in[1];
    const float*     Wl    = (const float*)d_in[2];
    const float*     bl    = (const float*)d_in[3];
    const float*     Wr    = (const float*)d_in[4];
    const float*     gamma = (const float*)d_in[5];
    const float*     beta  = (const float*)d_in[6];
    float*           out   = (float*)d_out;

    const int N = in_sizes[0] / FEAT;   // 50000
    const int E = in_sizes[1] / 2;      // 800000
    const int L = 3;

    const size_t nd_bytes = (size_t)N * FEAT * sizeof(float);
    char* ws = (char*)d_ws;
    float* aggb = (float*)(ws);                       // N*128 f32
    float* h0   = (float*)(ws + nd_bytes);            // N*128 f32
    float* h1   = (float*)(ws + 2 * nd_bytes);        // N*128 f32
    char*  p    = ws + 3 * nd_bytes;
    float* invc   = (float*)p;                 p += (size_t)N * 4;
    int*   deg    = (int*)p;                   p += (size_t)N * 4;   // deg+cursor contiguous
    int*   cursor = (int*)p;                   p += (size_t)N * 4;
    int*   offs   = (int*)p;                   p += (size_t)(N + 1) * 4;
    int*   srcs   = (int*)p;                   // E ints

    // ---- one-time CSR build (reused by all 3 layers) ----
    zero_i32<<<512, 256, 0, stream>>>(deg, 2L * N);               // deg + cursor
    degree_count<<<(E + 255) / 256, 256, 0, stream>>>(ei, deg, E);
    inv_count<<<(N + 255) / 256, 256, 0, stream>>>(deg, invc, N);
    exscan_deg<<<1, SCAN_T, 0, stream>>>(deg, offs, N);
    bucket_fill<<<(E + 255) / 256, 256, 0, stream>>>(ei, offs, cursor, srcs, E);
    sort_buckets<<<(N + 255) / 256, 256, 0, stream>>>(offs, srcs, N);

    const int tiles      = (N + 15) / 16;
    const int gemmBlocks = (tiles + 7) / 8;    // 8 waves / 256-thread block
    const int aggBlocks  = (N + 7) / 8;        // 1 wave / node

    const float* hcur = x;
    float* houts[3] = { h0, h1, out };
    for (int l = 0; l < L; ++l) {
        csr_aggregate<<<aggBlocks, 256, 0, stream>>>(offs, srcs, hcur, invc, aggb, N);
        gemm_ln_relu<<<gemmBlocks, 256, 0, stream>>>(
            aggb, hcur, invc,
            Wl + (size_t)l * FEAT * FEAT, bl + (size_t)l * FEAT,
            Wr + (size_t)l * FEAT * FEAT,
            gamma + (size_t)l * FEAT, beta + (size_t)l * FEAT,
            houts[l], N);
        hcur = houts[l];
    }
}